// InferenceEngine_82764019794611
// MI455X (gfx1250) — compile-verified
//
#include <hip/hip_runtime.h>
#include <math.h>
#include <limits.h>

#define Bn 128
#define Vn 128000
#define Tn 128
#define NB1 4096
#define NB2 4096
#define CHUNK 1024
#define NCHUNK (Vn / CHUNK)   /* 125 */
#define NBLK 5                /* strip blocks per row for strip kernels */
#define ITERS (NCHUNK / NBLK) /* 25 chunks per strip block, exact */
#define NTHREADS 256
#define HASHN 512
#define HASH_SHIFT 23         /* 32 - log2(HASHN) */

typedef __attribute__((ext_vector_type(2))) float v2f;
typedef __attribute__((ext_vector_type(8))) float v8f;

__device__ __forceinline__ float softplus_f(float x) {
  return fmaxf(x, 0.0f) + log1pf(__expf(-fabsf(x)));
}
__device__ __forceinline__ float sigmoid_f(float x) {
  return 1.0f / (1.0f + __expf(-x));
}

// 32-lane f32 sum via V_WMMA_F32_16X16X4_F32 with B = ones.
// A (16x4 f32): lane L<16 holds A[L][0..1], lane L+16 holds A[L][2..3].
// With a = {x, 0}: D[m][n] = x_m + x_{m+16}. Each lane's 8 D components cover
// 8 rows; summing them + xor-shuffle(16) yields the wave total in every lane.
// Caller must be at a full-EXEC point (uniform control flow, 256-thread block).
__device__ __forceinline__ float wave_sum_wmma(float x) {
  v2f a; a.x = x;    a.y = 0.0f;
  v2f b; b.x = 1.0f; b.y = 1.0f;
  v8f c = {0.f, 0.f, 0.f, 0.f, 0.f, 0.f, 0.f, 0.f};
  v8f d = __builtin_amdgcn_wmma_f32_16x16x4_f32(false, a, false, b, (short)0, c,
                                                false, false);
  float s = d[0] + d[1] + d[2] + d[3] + d[4] + d[5] + d[6] + d[7];
  s += __shfl_xor(s, 16, 32);
  return s;
}

// ---------------------------------------------------------------------------
// kA: penalty + temperature over a 25600-element strip; write z; per-strip
// (max, sumexp, min) partial via per-thread online softmax + WMMA reduce.
// ---------------------------------------------------------------------------
__global__ void __launch_bounds__(NTHREADS)
kA(const float* __restrict__ logits, const float* __restrict__ traw,
   const float* __restrict__ rraw, const int* __restrict__ ids,
   float* __restrict__ zout, float* __restrict__ partials /* [B][NBLK][4] */) {
  const int row = blockIdx.y;
  const int blk = blockIdx.x;
  const int tid = threadIdx.x;

  __shared__ int hkey[HASHN];
  __shared__ int hcnt[HASHN];
  __shared__ float redmax[8];
  __shared__ float redmin[8];
  __shared__ float redsum[8];
  __shared__ float blkM;

  for (int i = tid; i < HASHN; i += NTHREADS) { hkey[i] = -1; hcnt[i] = 0; }
  __syncthreads();
  if (tid < Tn) {
    int tok = ids[row * Tn + tid];
    unsigned slot = ((unsigned)tok * 2654435761u) >> HASH_SHIFT;
    while (true) {
      int prev = atomicCAS(&hkey[slot], -1, tok);
      if (prev == -1 || prev == tok) { atomicAdd(&hcnt[slot], 1); break; }
      slot = (slot + 1) & (HASHN - 1);
    }
  }
  __syncthreads();

  const float invT = 1.0f / softplus_f(traw[0]);
  const float lp = __logf(softplus_f(rraw[0]));

  float m_run  = -__builtin_inff();
  float s_run  = 0.f;
  float mn_run = __builtin_inff();

  for (int j = 0; j < ITERS; ++j) {
    const int vbase = (blk * ITERS + j) * CHUNK + tid * 4;
    const int base = row * Vn + vbase;
    float4 x = *(const float4*)(logits + base);
    float z[4];
    float xv4[4] = {x.x, x.y, x.z, x.w};
#pragma unroll
    for (int k = 0; k < 4; ++k) {
      float xv = xv4[k];
      int tok = vbase + k;
      int c = 0;
      unsigned slot = ((unsigned)tok * 2654435761u) >> HASH_SHIFT;
      while (true) {
        int kk = hkey[slot];
        if (kk == tok) { c = hcnt[slot]; break; }
        if (kk == -1) break;
        slot = (slot + 1) & (HASHN - 1);
      }
      float fac = (c > 0) ? __expf(lp * (xv > 0.0f ? -(float)c : (float)c)) : 1.0f;
      z[k] = xv * fac * invT;
    }
    *(float4*)(zout + base) = make_float4(z[0], z[1], z[2], z[3]);

    float m4 = fmaxf(fmaxf(z[0], z[1]), fmaxf(z[2], z[3]));
    mn_run = fminf(mn_run, fminf(fminf(z[0], z[1]), fminf(z[2], z[3])));
    float mnew = fmaxf(m_run, m4);
    float s4 = __expf(z[0] - mnew) + __expf(z[1] - mnew) +
               __expf(z[2] - mnew) + __expf(z[3] - mnew);
    s_run = s_run * __expf(m_run - mnew) + s4;   // first iter: 0 * exp(-inf) = 0
    m_run = mnew;
  }

  // block max/min
  float m = m_run, mn = mn_run;
#pragma unroll
  for (int off = 16; off > 0; off >>= 1) {
    m  = fmaxf(m,  __shfl_xor(m,  off, 32));
    mn = fminf(mn, __shfl_xor(mn, off, 32));
  }
  const int wave = tid >> 5;
  if ((tid & 31) == 0) { redmax[wave] = m; redmin[wave] = mn; }
  __syncthreads();
  if (tid == 0) {
    float M = redmax[0], Mn = redmin[0];
    for (int i = 1; i < 8; ++i) { M = fmaxf(M, redmax[i]); Mn = fminf(Mn, redmin[i]); }
    blkM = M; redmin[0] = Mn;
  }
  __syncthreads();
  const float M = blkM;

  float sadj = (s_run == 0.f) ? 0.f : s_run * __expf(m_run - M);
  float wsum = wave_sum_wmma(sadj);              // WMMA reduction, full EXEC
  if ((tid & 31) == 0) redsum[wave] = wsum;
  __syncthreads();
  if (tid == 0) {
    float S = 0.f;
    for (int i = 0; i < 8; ++i) S += redsum[i];
    ((float4*)partials)[row * NBLK + blk] = make_float4(M, S, redmin[0], 0.f);
  }
}

// ---------------------------------------------------------------------------
// kB: per-row combine of NBLK partials -> (M, S, min, range).
// ---------------------------------------------------------------------------
__global__ void __launch_bounds__(NTHREADS)
kB(const float* __restrict__ partials, float* __restrict__ rowStats) {
  const int row = blockIdx.x;
  const int tid = threadIdx.x;
  __shared__ float redmax[8];
  __shared__ float redmin[8];
  __shared__ float redsum[8];
  __shared__ float blkM, blkMn;

  float m = -__builtin_inff(), s = 0.f, mn = __builtin_inff();
  if (tid < NBLK) {
    float4 p = ((const float4*)partials)[row * NBLK + tid];
    m = p.x; s = p.y; mn = p.z;
  }
#pragma unroll
  for (int off = 16; off > 0; off >>= 1) {
    m  = fmaxf(m,  __shfl_xor(m,  off, 32));
    mn = fminf(mn, __shfl_xor(mn, off, 32));
  }
  const int wave = tid >> 5;
  if ((tid & 31) == 0) { redmax[wave] = m; redmin[wave] = mn; }
  __syncthreads();
  if (tid == 0) {
    float M = redmax[0], Mn = redmin[0];
    for (int i = 1; i < 8; ++i) { M = fmaxf(M, redmax[i]); Mn = fminf(Mn, redmin[i]); }
    blkM = M; blkMn = Mn;
  }
  __syncthreads();
  const float M = blkM;

  float sadj = (s == 0.f) ? 0.f : s * __expf(m - M);
  float wsum = wave_sum_wmma(sadj);              // WMMA reduction, full EXEC
  if ((tid & 31) == 0) redsum[wave] = wsum;
  __syncthreads();
  if (tid == 0) {
    float S = 0.f;
    for (int i = 0; i < 8; ++i) S += redsum[i];
    float* rs = rowStats + row * 16;
    rs[0] = M; rs[1] = S; rs[2] = blkMn; rs[3] = fmaxf(M - blkMn, 1e-20f);
  }
}

// ---------------------------------------------------------------------------
// kC: coarse exp-weighted histogram over a strip; private slice, plain stores.
// ---------------------------------------------------------------------------
__global__ void __launch_bounds__(NTHREADS)
kC(const float* __restrict__ z, const float* __restrict__ rowStats,
   float* __restrict__ hist1 /* [B][NBLK][NB1] */) {
  const int row = blockIdx.y, blk = blockIdx.x, tid = threadIdx.x;
  __shared__ float h[NB1];
  for (int i = tid; i < NB1; i += NTHREADS) h[i] = 0.f;
  __syncthreads();

  const float* rs = rowStats + row * 16;
  const float M = rs[0], mn = rs[2], rng = rs[3];
  const float inv1 = (float)NB1 / rng;

  for (int j = 0; j < ITERS; ++j) {
    const int base = row * Vn + (blk * ITERS + j) * CHUNK + tid * 4;
    float4 zz = *(const float4*)(z + base);
    float zv4[4] = {zz.x, zz.y, zz.z, zz.w};
#pragma unroll
    for (int k = 0; k < 4; ++k) {
      float zv = zv4[k];
      int b = (int)((zv - mn) * inv1);
      b = b < 0 ? 0 : (b > NB1 - 1 ? NB1 - 1 : b);
      atomicAdd(&h[b], __expf(zv - M));
    }
  }
  __syncthreads();
  float* dst = hist1 + ((size_t)(row * NBLK + blk)) * NB1;
  for (int i = tid; i < NB1; i += NTHREADS) dst[i] = h[i];
}

// ---------------------------------------------------------------------------
// kD: sum NBLK slices, suffix-scan from top, find critical coarse bin K1.
// ---------------------------------------------------------------------------
__global__ void __launch_bounds__(NTHREADS)
kD(const float* __restrict__ hist1, float* __restrict__ rowStats,
   const float* __restrict__ praw) {
  const int row = blockIdx.x, tid = threadIdx.x;
  __shared__ float csum[NTHREADS];
  __shared__ int kmin;

  float loc[16];
  const int base = tid * 16;
  float lsum = 0.f;
#pragma unroll
  for (int i = 0; i < 16; ++i) {
    float v = 0.f;
    for (int b = 0; b < NBLK; ++b)
      v += hist1[((size_t)(row * NBLK + b)) * NB1 + base + i];
    loc[i] = v; lsum += v;
  }
  csum[tid] = lsum;
  if (tid == 0) kmin = NB1 - 1;
  __syncthreads();
  if (tid == 0) {                      // exclusive suffix sums over 256 chunks
    float run = 0.f;
    for (int i = NTHREADS - 1; i >= 0; --i) { float t = csum[i]; csum[i] = run; run += t; }
  }
  __syncthreads();

  float* rs = rowStats + row * 16;
  const float P = sigmoid_f(praw[0]) * rs[1];

  float above = csum[tid];             // mass strictly above this chunk's top
  int cand = INT_MAX;
#pragma unroll
  for (int i = 15; i >= 0; --i) {
    if (above <= P) cand = base + i;
    above += loc[i];
  }
  if (cand != INT_MAX) atomicMin(&kmin, cand);
  __syncthreads();
  const int K1 = kmin;
  if (tid == (K1 >> 4)) {
    float ab = csum[tid];
    for (int i = 15; i > (K1 & 15); --i) ab += loc[i];   // == cumAbove(K1)
    const float mn = rs[2], rng = rs[3];
    rs[4] = (float)K1;
    rs[5] = ab;                                          // C1
    rs[6] = mn + (float)K1 * (rng / (float)NB1);         // lo1
  }
}

// ---------------------------------------------------------------------------
// kE: fine histogram restricted to critical coarse bin K1; private slices.
// ---------------------------------------------------------------------------
__global__ void __launch_bounds__(NTHREADS)
kE(const float* __restrict__ z, const float* __restrict__ rowStats,
   float* __restrict__ hist2 /* [B][NBLK][NB2] */) {
  const int row = blockIdx.y, blk = blockIdx.x, tid = threadIdx.x;
  __shared__ float h[NB2];
  for (int i = tid; i < NB2; i += NTHREADS) h[i] = 0.f;
  __syncthreads();

  const float* rs = rowStats + row * 16;
  const float M = rs[0], mn = rs[2], rng = rs[3];
  const int K1 = (int)rs[4];
  const float lo1 = rs[6];
  const float inv1 = (float)NB1 / rng;
  const float inv2 = (float)NB2 * (float)NB1 / rng;

  for (int j = 0; j < ITERS; ++j) {
    const int base = row * Vn + (blk * ITERS + j) * CHUNK + tid * 4;
    float4 zz = *(const float4*)(z + base);
    float zv4[4] = {zz.x, zz.y, zz.z, zz.w};
#pragma unroll
    for (int k = 0; k < 4; ++k) {
      float zv = zv4[k];
      int b1 = (int)((zv - mn) * inv1);
      b1 = b1 < 0 ? 0 : (b1 > NB1 - 1 ? NB1 - 1 : b1);
      if (b1 == K1) {
        int b2 = (int)((zv - lo1) * inv2);
        b2 = b2 < 0 ? 0 : (b2 > NB2 - 1 ? NB2 - 1 : b2);
        atomicAdd(&h[b2], __expf(zv - M));
      }
    }
  }
  __syncthreads();
  float* dst = hist2 + ((size_t)(row * NBLK + blk)) * NB2;
  for (int i = tid; i < NB2; i += NTHREADS) dst[i] = h[i];
}

// ---------------------------------------------------------------------------
// kF: sum slices, suffix-scan with base C1 -> threshold t*.
// ---------------------------------------------------------------------------
__global__ void __launch_bounds__(NTHREADS)
kF(const float* __restrict__ hist2, float* __restrict__ rowStats,
   const float* __restrict__ praw) {
  const int row = blockIdx.x, tid = threadIdx.x;
  __shared__ float csum[NTHREADS];
  __shared__ int kmin;

  float loc[16];
  const int base = tid * 16;
  float lsum = 0.f;
#pragma unroll
  for (int i = 0; i < 16; ++i) {
    float v = 0.f;
    for (int b = 0; b < NBLK; ++b)
      v += hist2[((size_t)(row * NBLK + b)) * NB2 + base + i];
    loc[i] = v; lsum += v;
  }
  csum[tid] = lsum;
  if (tid == 0) kmin = NB2 - 1;
  __syncthreads();
  if (tid == 0) {
    float run = 0.f;
    for (int i = NTHREADS - 1; i >= 0; --i) { float t = csum[i]; csum[i] = run; run += t; }
  }
  __syncthreads();

  float* rs = rowStats + row * 16;
  const float P = sigmoid_f(praw[0]) * rs[1];
  const float C1 = rs[5];

  float above = C1 + csum[tid];
  int cand = INT_MAX;
#pragma unroll
  for (int i = 15; i >= 0; --i) {
    if (above <= P) cand = base + i;
    above += loc[i];
  }
  if (cand != INT_MAX) atomicMin(&kmin, cand);
  __syncthreads();
  if (tid == 0) {
    const float lo1 = rs[6], rng = rs[3];
    rs[7] = lo1 + (float)kmin * (rng / ((float)NB1 * (float)NB2));  // t*
  }
}

// ---------------------------------------------------------------------------
// kG: final mask, in place on d_out (HBM-bound streaming pass, wide grid).
// ---------------------------------------------------------------------------
__global__ void __launch_bounds__(NTHREADS)
kG(float* __restrict__ out, const float* __restrict__ rowStats) {
  const int row = blockIdx.y, chunk = blockIdx.x, tid = threadIdx.x;
  const float t = rowStats[row * 16 + 7];
  const int base = row * Vn + chunk * CHUNK + tid * 4;
  float4 zz = *(const float4*)(out + base);
  const float ninf = -__builtin_inff();
  zz.x = (zz.x >= t) ? zz.x : ninf;
  zz.y = (zz.y >= t) ? zz.y : ninf;
  zz.z = (zz.z >= t) ? zz.z : ninf;
  zz.w = (zz.w >= t) ? zz.w : ninf;
  *(float4*)(out + base) = zz;
}

extern "C" void kernel_launch(void* const* d_in, const int* in_sizes, int n_in,
                              void* d_out, int out_size, void* d_ws, size_t ws_size,
                              hipStream_t stream) {
  (void)in_sizes; (void)n_in; (void)out_size; (void)ws_size;
  const float* logits = (const float*)d_in[0];
  const float* traw   = (const float*)d_in[1];
  const float* praw   = (const float*)d_in[2];
  const float* rraw   = (const float*)d_in[3];
  const int*   ids    = (const int*)d_in[4];
  float* out = (float*)d_out;

  float* ws       = (float*)d_ws;
  float* rowStats = ws;                                   // 128*16 f32
  float* partials = ws + Bn * 16;                         // 128*5*4 f32
  float* hist1    = partials + Bn * NBLK * 4;             // 128*5*4096 f32
  float* hist2    = hist1 + (size_t)Bn * NBLK * NB1;      // 128*5*4096 f32
  // total ~= 21 MB of workspace, all L2-resident

  dim3 strip(NBLK, Bn);     // 640 blocks, guard-free (125 = 5*25)
  dim3 wide(NCHUNK, Bn);    // 16000 blocks for HBM-bound passes
  kA<<<strip, NTHREADS, 0, stream>>>(logits, traw, rraw, ids, out, partials);
  kB<<<Bn,    NTHREADS, 0, stream>>>(partials, rowStats);
  kC<<<strip, NTHREADS, 0, stream>>>(out, rowStats, hist1);
  kD<<<Bn,    NTHREADS, 0, stream>>>(hist1, rowStats, praw);
  kE<<<strip, NTHREADS, 0, stream>>>(out, rowStats, hist2);
  kF<<<Bn,    NTHREADS, 0, stream>>>(hist2, rowStats, praw);
  kG<<<wide,  NTHREADS, 0, stream>>>(out, rowStats);
}